// GE2ELoss_6339371729741
// MI455X (gfx1250) — compile-verified
//
#include <hip/hip_runtime.h>

#define B_ROWS 32768
#define E_DIM  256
#define S_SPK  1024
#define EPS_COS  1e-8f
#define EPS_NORM 1e-12f

typedef float v2f __attribute__((ext_vector_type(2)));
typedef float v8f __attribute__((ext_vector_type(8)));

// ---------------- K0: zero segment accumulators + loss slot ----------------
__global__ void k0_zero(float* __restrict__ s, float* __restrict__ n,
                        float* __restrict__ loss_out) {
    int i = blockIdx.x * blockDim.x + threadIdx.x;
    if (i < S_SPK) { s[i] = 0.0f; n[i] = 0.0f; }
    if (i == 0) *loss_out = 0.0f;
}

// ---------------- K1: per-row sum & sum-of-squares via f32 WMMA ------------
// One wave handles a 16-row tile. A-matrix (16x4 f32) layout per ISA 7.12.2:
//   lanes 0-15:  M=lane, VGPR0=K0, VGPR1=K1 ;  lanes 16-31: M=lane-16, K2/K3.
// B = ones => D[m,n] = sum_k A[m,k] + C[m,n]  (B layout immaterial).
// C/D layout: lane<16 VGPR r = D[r, lane]; lane>=16 VGPR r = D[8+r, lane-16],
// and all N columns are identical, so lane 0 holds rows 0..7, lane 16 rows 8..15.
__global__ void __launch_bounds__(256)
k1_rowstats(const float* __restrict__ X, const int* __restrict__ targets,
            float* __restrict__ row_sum, float* __restrict__ norm_e,
            float* __restrict__ s, float* __restrict__ n) {
    const int lane  = threadIdx.x & 31;
    const int wv    = threadIdx.x >> 5;
    const int row0  = blockIdx.x * 128 + wv * 16;      // 8 waves * 16 rows
    const int m     = lane & 15;
    const int khalf = (lane >> 4) << 1;                // 0 or 2
    const float* rowp = X + (size_t)(row0 + m) * E_DIM + khalf;

    v8f csum = {0.f,0.f,0.f,0.f,0.f,0.f,0.f,0.f};
    v8f csq  = {0.f,0.f,0.f,0.f,0.f,0.f,0.f,0.f};
    const v2f ones = {1.0f, 1.0f};

    #pragma unroll 8
    for (int k0 = 0; k0 < E_DIM; k0 += 4) {
        v2f a  = *(const v2f*)(rowp + k0);   // 8B-aligned b64 load
        v2f a2 = a * a;
        csum = __builtin_amdgcn_wmma_f32_16x16x4_f32(
                   false, a,  false, ones, (short)0, csum, false, false);
        csq  = __builtin_amdgcn_wmma_f32_16x16x4_f32(
                   false, a2, false, ones, (short)0, csq,  false, false);
    }

    if ((lane & 15) == 0) {                 // lanes 0 and 16
        const int rbase = row0 + (lane >> 4) * 8;
        #pragma unroll
        for (int i = 0; i < 8; ++i) {
            const float rs = csum[i];
            const float sq = csq[i];
            const int   r  = rbase + i;
            row_sum[r] = rs;
            norm_e[r]  = sqrtf(sq);
            const int t = targets[r];
            atomicAdd(&s[t], rs);
            atomicAdd(&n[t], 1.0f);
        }
    }
}

// ---------------- K2: per-speaker scalar centroid factor -------------------
// c_k = s_k/n_k ; norm_ck = |c_k|*sqrt(E)=|c_k|*16 ; g_k = c_k/max(norm_ck,eps)
__global__ void k2_speaker(const float* __restrict__ s, const float* __restrict__ n,
                           float* __restrict__ g) {
    int k = blockIdx.x * blockDim.x + threadIdx.x;
    if (k < S_SPK) {
        float c = s[k] / n[k];
        g[k] = c / fmaxf(fabsf(c) * 16.0f, EPS_COS);
    }
}

// ---------------- K3: closed-form self-sim + rank-1 logsumexp + loss -------
__global__ void __launch_bounds__(256)
k3_loss(const float* __restrict__ row_sum, const float* __restrict__ norm_e,
        const int* __restrict__ targets, const float* __restrict__ s,
        const float* __restrict__ n, const float* __restrict__ g,
        const float* __restrict__ wp, const float* __restrict__ bp,
        float* __restrict__ loss_out) {
    __shared__ float gl[S_SPK];
    __shared__ float red[256];
    for (int i = threadIdx.x; i < S_SPK; i += 256) gl[i] = g[i];
    __syncthreads();

    const int   r    = blockIdx.x * 256 + threadIdx.x;
    const float wr   = fmaxf(wp[0], 0.0f);
    const float bias = bp[0];

    const float rs = row_sum[r];
    const float ne = norm_e[r];
    const int   t  = targets[r];
    const float sj = s[t];
    const float nj = n[t];

    const float sumsq    = ne * ne;
    const float dot      = (sj * rs - sumsq) / nj;
    const float normc2   = ((float)E_DIM * sj * sj - 2.0f * sj * rs + sumsq) / (nj * nj);
    const float normc    = sqrtf(normc2);
    const float self_cos = dot / (fmaxf(ne, EPS_COS) * fmaxf(normc, EPS_COS));
    const float ut       = wr * self_cos;        // self_sim - bias
    const float self_sim = ut + bias;

    const float a = wr * rs / fmaxf(ne, EPS_COS);
    // |g_k| <= 1/16, so C >= max over all entries: exact-safe logsumexp shift
    const float C = fmaxf(fabsf(a) * 0.0625f, ut);

    float sum = 0.0f;
    #pragma unroll 4
    for (int k = 0; k < S_SPK; ++k)
        sum += __expf(a * gl[k] - C);            // LDS broadcast read
    // replace own slot with self-sim (branch-free correction)
    sum += __expf(ut - C) - __expf(a * gl[t] - C);

    const float lse  = bias + C + __logf(sum);
    float term = -self_sim + lse;

    red[threadIdx.x] = term;
    __syncthreads();
    #pragma unroll
    for (int off = 128; off > 0; off >>= 1) {
        if (threadIdx.x < off) red[threadIdx.x] += red[threadIdx.x + off];
        __syncthreads();
    }
    if (threadIdx.x == 0) atomicAdd(loss_out, red[0]);
}

// ---------------- K4: normalized output, b128 in / b128 out ----------------
__global__ void __launch_bounds__(256)
k4_normalize(const float* __restrict__ X, const float* __restrict__ norm_e,
             float* __restrict__ out) {
    const size_t i    = (size_t)blockIdx.x * 256 + threadIdx.x;
    const size_t base = i * 4;                    // 4 floats per thread
    const int    r    = (int)(base >> 8);         // /E_DIM (256|4 -> same row)
    const float  inv  = 1.0f / fmaxf(norm_e[r], EPS_NORM);
    float4 x = *(const float4*)(X + base);
    float4 o;
    o.x = x.x * inv; o.y = x.y * inv; o.z = x.z * inv; o.w = x.w * inv;
    *(float4*)(out + base) = o;
}

// ---------------------------------------------------------------------------
extern "C" void kernel_launch(void* const* d_in, const int* in_sizes, int n_in,
                              void* d_out, int out_size, void* d_ws, size_t ws_size,
                              hipStream_t stream) {
    const float* X  = (const float*)d_in[0];   // [B,E] fp32
    const int*   tg = (const int*)d_in[1];     // [B] int32
    const float* wp = (const float*)d_in[2];   // scalar
    const float* bp = (const float*)d_in[3];   // scalar

    float* out = (float*)d_out;                // [B*E] normalized, then loss
    float* ws  = (float*)d_ws;
    float* row_sum = ws;                       // B floats
    float* norm_e  = ws + B_ROWS;              // B floats
    float* s       = ws + 2 * B_ROWS;          // S floats
    float* n       = s + S_SPK;                // S floats
    float* g       = n + S_SPK;                // S floats
    float* loss_out = out + (size_t)B_ROWS * E_DIM;

    k0_zero<<<4, 256, 0, stream>>>(s, n, loss_out);
    k1_rowstats<<<B_ROWS / 128, 256, 0, stream>>>(X, tg, row_sum, norm_e, s, n);
    k2_speaker<<<S_SPK / 256, 256, 0, stream>>>(s, n, g);
    k3_loss<<<B_ROWS / 256, 256, 0, stream>>>(row_sum, norm_e, tg, s, n, g, wp, bp, loss_out);
    k4_normalize<<<(B_ROWS * E_DIM / 4) / 256, 256, 0, stream>>>(X, norm_e, out);
}